// YOLOLossPerFeatureMapV2_84370337563356
// MI455X (gfx1250) — compile-verified
//
#include <hip/hip_runtime.h>
#include <math.h>

// ---------------- problem constants ----------------
#define NUM_CLASSES 80
#define GH 80
#define GW 80
#define NA 3
#define NBATCH 16
#define NANN 64
#define HW (GH * GW)                 // 6400
#define CELLS (NBATCH * NA * HW)     // 307200
#define BLK 256
#define NBLOCKS (CELLS / BLK)        // 1200 (divisible by 4 for WMMA K=4 steps)
#define BLOCKS_PER_PLANE (HW / BLK)  // 25
#define SENTINEL 0x7FFFFFFF
#define NCH 10                       // live channels: obj, bbox[3], cls[3], cnt[3]
#define PCH 16                       // padded channels (16 WMMA columns; 10..15 are zero)

typedef __attribute__((ext_vector_type(2))) float v2f;
typedef __attribute__((ext_vector_type(8))) float v8f;

// ---------------- helpers ----------------
__device__ __forceinline__ float bce(float x, float y) {
    // max(x,0) - x*y + log1p(exp(-|x|))
    return fmaxf(x, 0.0f) - x * y + log1pf(expf(-fabsf(x)));
}
__device__ __forceinline__ float wave_red(float v) {
#pragma unroll
    for (int off = 16; off > 0; off >>= 1) v += __shfl_down(v, off, 32);
    return v;
}

// ---------------- kernel 1: init winner array ----------------
__global__ void yolo_init_kernel(int* __restrict__ winner) {
    int i = blockIdx.x * BLK + threadIdx.x;
    if (i < CELLS) winner[i] = SENTINEL;
}

// ---------------- kernel 2: anchor matching (segment_min via atomicMin) ----------------
__global__ void yolo_match_kernel(const float* __restrict__ ann,
                                  const float* __restrict__ anchors,
                                  int* __restrict__ winner) {
    int b = blockIdx.x;          // batch
    int p = threadIdx.x;         // 0..191  (p = ai*NA + ki)
    if (p >= NANN * NA) return;
    int ai = p / NA;
    int ki = p % NA;
    const float* a = ann + (b * NANN + ai) * 5;
    float bw = a[2], bh = a[3];
    float aw = anchors[ki * 4 + 2];
    float ah = anchors[ki * 4 + 3];
    const float g2 = 4.0f;       // ANCHOR_GAIN^2
    const float epsm = 1.0e-4f;  // EPS_MATCH
    bool valid = (aw * g2 - epsm >= bw) && (aw / g2 + epsm <= bw) &&
                 (ah * g2 - epsm >= bh) && (ah / g2 + epsm <= bh);
    if (valid) {
        float gx = a[0] * (float)GW;
        float gy = a[1] * (float)GH;
        int gxi = (int)gx;
        int gyi = (int)gy;
        int lin = ki * HW + gyi * GW + gxi;
        atomicMin(&winner[b * (NA * HW) + lin], p);
    }
}

// ---------------- kernel 3: per-cell loss + block reduction to partials ----------------
__global__ void yolo_loss_kernel(const float* __restrict__ preds,
                                 const float* __restrict__ ann,
                                 const float* __restrict__ anchors,
                                 const int* __restrict__ winner,
                                 float* __restrict__ partials) {
    const int cell = blockIdx.x * BLK + threadIdx.x;
    const int x = cell % GW;
    const int y = (cell / GW) % GH;
    const int k = (cell / HW) % NA;
    const int b = cell / (NA * HW);

    // predictions layout: [b][k][c][y][x], channel stride = HW
    const float* pp = preds + ((b * NA + k) * (5 + NUM_CLASSES)) * HW + y * GW + x;

    const float p4 = pp[4 * HW];

    const int w = winner[cell];
    const bool masked = (w != SENTINEL);

    float v_obj, v_box = 0.0f, v_cls = 0.0f, v_cnt = 0.0f;

    if (masked) {
        const float p0 = pp[0];
        const float p1 = pp[1 * HW];
        const float p2 = pp[2 * HW];
        const float p3 = pp[3 * HW];

        const int ai = w / NA;
        const float* a = ann + (b * NANN + ai) * 5;
        const float gx = a[0] * (float)GW;
        const float gy = a[1] * (float)GH;
        const float tx = gx - (float)((int)gx);
        const float ty = gy - (float)((int)gy);
        const float tw = a[2] * (float)GW;
        const float th = a[3] * (float)GH;
        const int cls = (int)a[4];

        const float aw = anchors[k * 4 + 2];
        const float ah = anchors[k * 4 + 3];

        // decode predicted box (ANCHOR_GAIN = 2, half = 0.5)
        const float sx = 1.0f / (1.0f + expf(-p0));
        const float sy = 1.0f / (1.0f + expf(-p1));
        const float sw = 1.0f / (1.0f + expf(-p2));
        const float sh = 1.0f / (1.0f + expf(-p3));
        const float px = sx * 2.0f - 0.5f;
        const float py = sy * 2.0f - 0.5f;
        const float pw = (sw * 2.0f) * (sw * 2.0f) * aw * (float)GW;
        const float ph = (sh * 2.0f) * (sh * 2.0f) * ah * (float)GH;

        // CIoU
        const float eps = 1.0e-7f;
        const float pw2 = pw * 0.5f, ph2 = ph * 0.5f;
        const float tw2 = tw * 0.5f, th2 = th * 0.5f;
        float iw = fminf(px + pw2, tx + tw2) - fmaxf(px - pw2, tx - tw2);
        float ih = fminf(py + ph2, ty + th2) - fmaxf(py - ph2, ty - th2);
        iw = fmaxf(iw, 0.0f);
        ih = fmaxf(ih, 0.0f);
        const float inter = iw * ih;
        const float uni = pw * ph + tw * th - inter + eps;
        const float iou = inter / uni;
        const float cw = fmaxf(px + pw2, tx + tw2) - fminf(px - pw2, tx - tw2);
        const float chh = fmaxf(py + ph2, ty + th2) - fminf(py - ph2, ty - th2);
        const float c2 = cw * cw + chh * chh + eps;
        const float rho2 = (px - tx) * (px - tx) + (py - ty) * (py - ty);
        const float dat = atanf(tw / (th + eps)) - atanf(pw / (ph + eps));
        const float v = (float)(4.0 / (M_PI * M_PI)) * dat * dat;
        const float alpha = v / (v - iou + 1.0f + eps);
        const float ciou = iou - (rho2 / c2 + alpha * v);

        v_box = 1.0f - ciou;
        v_cnt = 1.0f;

        const float obj_t = fminf(fmaxf(iou, 0.0f), 1.0f);
        v_obj = bce(p4, obj_t);

        // class BCE (target one-hot at cls)
        float cacc = 0.0f;
        const float* pc = pp + 5 * HW;
#pragma unroll 4
        for (int c = 0; c < NUM_CLASSES; ++c) {
            const float xc = pc[c * HW];
            const float yc = (c == cls) ? 1.0f : 0.0f;
            cacc += bce(xc, yc);
        }
        v_cls = cacc;
    } else {
        v_obj = bce(p4, 0.0f);
    }

    // block reduction (8 waves of 32)
    __shared__ float red[BLK / 32][4];
    const float r0 = wave_red(v_obj);
    const float r1 = wave_red(v_box);
    const float r2 = wave_red(v_cls);
    const float r3 = wave_red(v_cnt);
    const int wave = threadIdx.x >> 5;
    const int lane = threadIdx.x & 31;
    if (lane == 0) {
        red[wave][0] = r0; red[wave][1] = r1; red[wave][2] = r2; red[wave][3] = r3;
    }
    __syncthreads();
    if (threadIdx.x == 0) {
        float so = 0.0f, sb = 0.0f, sc = 0.0f, sn = 0.0f;
#pragma unroll
        for (int ww = 0; ww < BLK / 32; ++ww) {
            so += red[ww][0]; sb += red[ww][1]; sc += red[ww][2]; sn += red[ww][3];
        }
        const int bid = blockIdx.x;
        const int kb = (blockIdx.x / BLOCKS_PER_PLANE) % NA;  // whole block shares one anchor k
        partials[0 * NBLOCKS + bid] = so;
#pragma unroll
        for (int kk = 0; kk < NA; ++kk) {
            partials[(1 + kk) * NBLOCKS + bid] = (kk == kb) ? sb : 0.0f;
            partials[(4 + kk) * NBLOCKS + bid] = (kk == kb) ? sc : 0.0f;
            partials[(7 + kk) * NBLOCKS + bid] = (kk == kb) ? sn : 0.0f;
        }
        // zero the padded WMMA columns (channels 10..15)
#pragma unroll
        for (int cc = NCH; cc < PCH; ++cc) {
            partials[cc * NBLOCKS + bid] = 0.0f;
        }
    }
}

// ---------------- kernel 4: WMMA column-sum final reduction ----------------
// D = ones(16x4) x B(4x16) + C : every row of D = per-column sums of B.
// Column n <- partial channel n (n=10..15 padded zero). Lane L feeds column
// L%16; per iteration each lane supplies 2 of the 4 K-slots (K = vgpr + 2*(L/16)),
// so one float2 load (global_load_b64) covers both VGPRs of B per lane.
__global__ void yolo_final_kernel(const float* __restrict__ partials,
                                  float* __restrict__ out) {
    const int lane = threadIdx.x;   // 0..31, one full wave, EXEC all ones
    const int ch = lane & 15;
    const int g = lane >> 4;

    v2f aones; aones.x = 1.0f; aones.y = 1.0f;
    v8f acc = {};

    const float* col = partials + ch * NBLOCKS + 2 * g;

    for (int i = 0; i < NBLOCKS; i += 4) {
        const v2f bvec = *(const v2f*)(col + i);   // unconditional b64 load
        acc = __builtin_amdgcn_wmma_f32_16x16x4_f32(
            /*neg_a=*/false, aones, /*neg_b=*/false, bvec,
            /*c_mod=*/(short)0, acc, /*reuse_a=*/false, /*reuse_b=*/false);
    }

    // acc[0] on lane n (n<16) = D[M=0][N=n] = total sum of channel n
    const float s = acc[0];

    const float obj_sum = __shfl(s, 0, 32);
    float loss = obj_sum / (float)CELLS;   // obj BCE mean
#pragma unroll
    for (int kk = 0; kk < NA; ++kk) {
        const float bbox = __shfl(s, 1 + kk, 32);
        const float clsv = __shfl(s, 4 + kk, 32);
        const float cnt  = __shfl(s, 7 + kk, 32);
        if (cnt > 0.0f) {
            loss += bbox / fmaxf(cnt, 1.0f);
            loss += clsv / fmaxf(cnt * (float)NUM_CLASSES, 1.0f);
        }
    }
    if (lane == 0) out[0] = loss;
}

// ---------------- launch ----------------
extern "C" void kernel_launch(void* const* d_in, const int* in_sizes, int n_in,
                              void* d_out, int out_size, void* d_ws, size_t ws_size,
                              hipStream_t stream) {
    const float* preds   = (const float*)d_in[0];   // (16,3,85,80,80)
    const float* ann     = (const float*)d_in[1];   // (16,64,5)
    const float* anchors = (const float*)d_in[2];   // (3,4)
    float* out = (float*)d_out;

    // workspace layout: [winner: CELLS ints][partials: PCH*NBLOCKS floats]
    int*   winner   = (int*)d_ws;
    float* partials = (float*)((char*)d_ws + (size_t)CELLS * sizeof(int));

    yolo_init_kernel<<<NBLOCKS, BLK, 0, stream>>>(winner);
    yolo_match_kernel<<<NBATCH, NANN * NA, 0, stream>>>(ann, anchors, winner);
    yolo_loss_kernel<<<NBLOCKS, BLK, 0, stream>>>(preds, ann, anchors, winner, partials);
    yolo_final_kernel<<<1, 32, 0, stream>>>(partials, out);
}